// GATModel_7318624273063
// MI455X (gfx1250) — compile-verified
//
#include <hip/hip_runtime.h>
#include <stdint.h>

// Problem constants (from reference)
#define B_   4
#define N_   2048
#define FIN  64
#define HID  128
#define H_   4
#define D_   32
#define NEG_SLOPE 0.2f

typedef float v2f __attribute__((ext_vector_type(2)));
typedef float v8f __attribute__((ext_vector_type(8)));

// ---------------------------------------------------------------------------
// GEMM: Out[M x 128] = Xin[M x K] @ W[K x 128] (+bias), fp32 WMMA 16x16x4.
// One wave computes one 16x16 output tile; 8 waves/block cover all 128 cols
// of a 16-row strip. X strip staged in padded LDS. W (<=64KB) is L2/L0
// resident, read directly. (Codegen verified: chained v_wmma_f32_16x16x4_f32
// with ds_load_2addr_b64 A-fragments.)
// ---------------------------------------------------------------------------
template <int K>
__global__ __launch_bounds__(256) void gemm_wmma_f32(
    const float* __restrict__ Xin, const float* __restrict__ W,
    const float* __restrict__ bias, float* __restrict__ Out)
{
    __shared__ float As[16][K + 4];       // +4 pad: kill LDS bank conflicts
    const int m0  = blockIdx.x * 16;
    const int tid = threadIdx.x;

    const float* src = Xin + (size_t)m0 * K;      // 16-row strip is contiguous
    for (int e = tid; e < 16 * K; e += 256) As[e / K][e % K] = src[e];
    __syncthreads();

    const int wave = tid >> 5, lane = tid & 31;
    const int lo = lane & 15, hi = lane >> 4;
    const int n0 = wave * 16;

    v8f acc = {};
#pragma unroll
    for (int kk = 0; kk < K; kk += 4) {
        const int kb = kk + 2 * hi;
        v2f a, b;
        a[0] = As[lo][kb];
        a[1] = As[lo][kb + 1];
        b[0] = W[(size_t)kb * HID + n0 + lo];
        b[1] = W[(size_t)(kb + 1) * HID + n0 + lo];
        acc = __builtin_amdgcn_wmma_f32_16x16x4_f32(
            false, a, false, b, (short)0, acc, false, false);
    }

    const float bv = bias ? bias[n0 + lo] : 0.0f;
#pragma unroll
    for (int v = 0; v < 8; ++v)
        Out[(size_t)(m0 + v + 8 * hi) * HID + n0 + lo] = acc[v] + bv;
}

// ---------------------------------------------------------------------------
// Pack adjacency to bitmask: Abits[row][jw] bit l = (A[row][jw*32+l] != 0).
// One wave per row; wave32 ballot produces one u32 word per 32 columns.
// One streaming pass over A (67 MB @ 23.3 TB/s ~ 3us); layers then read 2 MB.
// ---------------------------------------------------------------------------
__global__ __launch_bounds__(256) void pack_kernel(
    const int* __restrict__ A, uint32_t* __restrict__ Abits)
{
    const int wave = threadIdx.x >> 5, lane = threadIdx.x & 31;
    const int row = blockIdx.x * 8 + wave;        // b*N + i
    const int* Arow = A + (size_t)row * N_;
    uint32_t* orow = Abits + (size_t)row * (N_ / 32);
    for (int jw = 0; jw < N_ / 32; ++jw) {
        const int a = Arow[jw * 32 + lane];       // coalesced b32
        const unsigned long long m = __ballot(a != 0);
        if (lane == 0) orow[jw] = (uint32_t)m;
    }
}

// ---------------------------------------------------------------------------
// Per-node scores + factored exponential table.
// One wave per node; lane == d (D == 32 == wave32 width).
//   s_arr[(b*H+h)*N + n] = <Xs[b,n,h,:], a_src[h,:]>
//   F[(b*H+h)*N + n]     = ( exp(t), exp(0.2*t) )   with t = <Xs, a_dst>
// (t itself is not stored: sign test s+t>=0 <=> exp(t) >= exp(-s).)
// ---------------------------------------------------------------------------
__global__ __launch_bounds__(256) void st_kernel(
    const float* __restrict__ Xs, const float* __restrict__ a_src,
    const float* __restrict__ a_dst, float* __restrict__ s_arr,
    float2* __restrict__ F)
{
    const int wave = threadIdx.x >> 5, lane = threadIdx.x & 31;
    const int node = blockIdx.x * 8 + wave;   // b*N + n
    const int b = node / N_, n = node % N_;
    const float* xrow = Xs + (size_t)node * HID;

#pragma unroll
    for (int h = 0; h < H_; ++h) {
        const float v  = xrow[h * D_ + lane];
        float sd = v * a_src[h * D_ + lane];
        float td = v * a_dst[h * D_ + lane];
#pragma unroll
        for (int off = 16; off > 0; off >>= 1) {
            sd += __shfl_xor(sd, off, 32);
            td += __shfl_xor(td, off, 32);
        }
        if (lane == 0) {
            const size_t o = (size_t)(b * H_ + h) * N_ + n;
            s_arr[o] = sd;
            F[o] = make_float2(__expf(td), __expf(NEG_SLOPE * td));
        }
    }
}

// ---------------------------------------------------------------------------
// Diagonal-softmax attention. One wave per row (b,i); 8 rows (same batch) per
// block. The batch's 64 KB (e^t, e^{0.2t}) table is staged in LDS once and
// shared by all 8 waves. Adjacency comes from the packed bitmask: one uniform
// u32 per 32 neighbors. Hot loop has zero transcendentals and zero per-lane
// global loads.
// ---------------------------------------------------------------------------
__global__ __launch_bounds__(256) void attn_kernel(
    const uint32_t* __restrict__ Abits, const float* __restrict__ s_arr,
    const float2* __restrict__ F, const float* __restrict__ Xs,
    float* __restrict__ Out)
{
    __shared__ float2 Fs[H_ * N_];                // 64 KB
    const int tid = threadIdx.x;
    const int wave = tid >> 5, lane = tid & 31;
    const int b = blockIdx.x >> 8;                // N_/8 = 256 blocks per batch
    const int i = ((blockIdx.x & 255) << 3) + wave;
    const int row = b * N_ + i;

    // Stage batch's exp-table: 64 KB, b128 global -> b128 LDS, 16 iters/thread
    {
        const float4* src = (const float4*)(F + (size_t)b * H_ * N_);
        float4* dst = (float4*)Fs;
        for (int e = tid; e < (H_ * N_) / 2; e += 256) dst[e] = src[e];
    }
    __syncthreads();

    float P1[H_], P2[H_], T[H_], acc[H_];
#pragma unroll
    for (int h = 0; h < H_; ++h) {
        const float s = s_arr[(size_t)(b * H_ + h) * N_ + i];
        P1[h] = __expf(s);                        // e^{s}
        P2[h] = __expf(NEG_SLOPE * s);            // e^{0.2 s}
        T[h]  = __expf(-s);                       // threshold: e^t >= e^{-s}
        acc[h] = 0.0f;
    }

    const uint32_t* wrow = Abits + (size_t)row * (N_ / 32);
    for (int jw = 0; jw < N_ / 32; jw += 2) {
        const uint2 wp = *(const uint2*)(wrow + jw);   // uniform 8B load
        int j = jw * 32 + lane;
#pragma unroll
        for (int half = 0; half < 2; ++half) {
            const uint32_t w = half ? wp.y : wp.x;
            const uint32_t bit = (w >> lane) & 1u;
#pragma unroll
            for (int h = 0; h < H_; ++h) {
                const float2 f = Fs[h * N_ + j];       // ds_load_b64
                const float e = (f.x >= T[h]) ? (P1[h] * f.x) : (P2[h] * f.y);
                acc[h] += bit ? e : 0.0f;
            }
            j += 32;
        }
    }

    // wave32 butterfly: every lane ends with the full denominator
#pragma unroll
    for (int h = 0; h < H_; ++h) {
#pragma unroll
        for (int off = 16; off > 0; off >>= 1)
            acc[h] += __shfl_xor(acc[h], off, 32);
    }

    // diagonal numerator + scaled writeback
    const float mii = (float)((wrow[i >> 5] >> (i & 31)) & 1u);
    const float* xrow = Xs + (size_t)row * HID;
    float* orow = Out + (size_t)row * HID;
#pragma unroll
    for (int h = 0; h < H_; ++h) {
        const float2 f = Fs[h * N_ + i];
        const float num = mii * ((f.x >= T[h]) ? (P1[h] * f.x) : (P2[h] * f.y));
        const float diag = num / acc[h];
        orow[h * D_ + lane] = diag * xrow[h * D_ + lane];
    }
}

// ---------------------------------------------------------------------------
extern "C" void kernel_launch(void* const* d_in, const int* in_sizes, int n_in,
                              void* d_out, int out_size, void* d_ws, size_t ws_size,
                              hipStream_t stream)
{
    const float* X     = (const float*)d_in[0];   // (B,N,64)
    const int*   A     = (const int*)  d_in[1];   // (B,N,N)
    const float* W_emb = (const float*)d_in[2];   // (64,128)
    const float* b_emb = (const float*)d_in[3];   // (128,)
    const float* Wl[3]   = {(const float*)d_in[4],  (const float*)d_in[7],  (const float*)d_in[10]};
    const float* asrc[3] = {(const float*)d_in[5],  (const float*)d_in[8],  (const float*)d_in[11]};
    const float* adst[3] = {(const float*)d_in[6],  (const float*)d_in[9],  (const float*)d_in[12]};

    // Workspace layout (~10.4 MB total)
    float*    buf0  = (float*)d_ws;                               // 4 MB
    float*    buf1  = buf0 + (size_t)B_ * N_ * HID;               // 4 MB
    float*    s_arr = buf1 + (size_t)B_ * N_ * HID;               // 128 KB
    float2*   F     = (float2*)(s_arr + (size_t)B_ * H_ * N_);    // 256 KB
    uint32_t* Abits = (uint32_t*)(F + (size_t)B_ * H_ * N_);      // 2 MB

    const int rows = B_ * N_;                                     // 8192

    // One-time: pack adjacency to bits (A read exactly once from HBM)
    pack_kernel<<<rows / 8, 256, 0, stream>>>(A, Abits);

    // Embedding GEMM (K=64) with bias
    gemm_wmma_f32<FIN><<<rows / 16, 256, 0, stream>>>(X, W_emb, b_emb, buf0);

    for (int l = 0; l < 3; ++l) {
        gemm_wmma_f32<HID><<<rows / 16, 256, 0, stream>>>(buf0, Wl[l], nullptr, buf1);
        st_kernel<<<rows / 8, 256, 0, stream>>>(buf1, asrc[l], adst[l], s_arr, F);
        float* out_l = (l == 2) ? (float*)d_out : buf0;
        attn_kernel<<<rows / 8, 256, 0, stream>>>(Abits, s_arr, F, buf1, out_l);
    }
}